// HyperedgeAggregator_36670430773458
// MI455X (gfx1250) — compile-verified
//
#include <hip/hip_runtime.h>
#include <hip/hip_bf16.h>
#include <math.h>

// Problem constants (fixed by the reference)
#define B_ 2
#define N_ 20000
#define E_ 20000
#define A_ 8
#define H_ 256
#define TE_ 32
#define KENC 288   // H + TE
#define KUPD 512   // 2*H

#define GLOBAL_AS __attribute__((address_space(1)))
#define LDS_AS    __attribute__((address_space(3)))

typedef __attribute__((ext_vector_type(2))) float v2f;
typedef __attribute__((ext_vector_type(8))) float v8f;
typedef int v4i __attribute__((vector_size(16)));   // matches builtin param type

// ---------------------------------------------------------------------------
// Kernel 1: masked mean-pool of member node features + edge-type embedding.
// Writes the encoder input panel xin[B*E, 288]  (cols 0..255 pooled, 256..287 emb)
// ---------------------------------------------------------------------------
__global__ __launch_bounds__(256)
void pool_kernel(const float* __restrict__ nf,          // [B,N,H]
                 const int* __restrict__ members,       // [B,E,A]
                 const int* __restrict__ types,         // [B,E]
                 const unsigned char* __restrict__ mask,// [B,E,A] (bool)
                 const float* __restrict__ ett,         // [NT,TE]
                 float* __restrict__ xin)                // [B*E, 288]
{
    const int be = blockIdx.x;            // b*E + e
    const int b  = be / E_;
    const int h  = threadIdx.x;           // 0..255
    const int* mem = members + (size_t)be * A_;
    const unsigned char* mk = mask + (size_t)be * A_;

    float cnt = 0.f, acc = 0.f;
#pragma unroll
    for (int a = 0; a < A_; ++a) {
        int idx = mem[a];
        idx = idx < 0 ? 0 : (idx > N_ - 1 ? N_ - 1 : idx);
        float m = mk[a] ? 1.f : 0.f;
        cnt += m;
        acc += m * nf[((size_t)b * N_ + idx) * H_ + h];
    }
    float pooled = acc / fmaxf(cnt, 1.f);
    xin[(size_t)be * KENC + h] = pooled;
    if (h < TE_) {
        int t = types[be];
        xin[(size_t)be * KENC + H_ + h] = ett[t * TE_ + h];
    }
}

// ---------------------------------------------------------------------------
// WMMA fp32 GEMM + bias + exact GELU + LayerNorm, fused per 16x256 tile.
// Block = 512 threads = 16 wave32s.  Wave w computes the 16x16 tile at
// N-range [16w, 16w+16), then LayerNorms row w of the block tile.
// A-panel staged into LDS via GLOBAL_LOAD_ASYNC_TO_LDS_B128 when available.
// SPLIT: A row = concat(A0[row,0:256], A1[row,0:256])  (for the update GEMM)
// ---------------------------------------------------------------------------
template<int K, bool SPLIT>
__global__ __launch_bounds__(512)
void wmma_gemm_gelu_ln(const float* __restrict__ A0,
                       const float* __restrict__ A1,
                       const float* __restrict__ W,      // [K,256] row-major
                       const float* __restrict__ bias,   // [256]
                       const float* __restrict__ gamma,  // [256]
                       const float* __restrict__ beta,   // [256]
                       float* __restrict__ out)          // [rows,256]
{
    // PAD=4 so each LDS row stride (K+4 floats) is a multiple of 16 bytes:
    // async b128 chunk destinations stay 16B aligned for every row.
    constexpr int PAD = 4;
    __shared__ float As[16][K + PAD];
    __shared__ float sm[16][260];

    const int row0 = blockIdx.x * 16;
    const int tid  = threadIdx.x;

    // ---- Stage the 16 x K A-panel into LDS ----
#if __has_builtin(__builtin_amdgcn_global_load_async_to_lds_b128)
    {
        constexpr int CPR = K / 4;            // 16-byte chunks per row
        for (int c = tid; c < 16 * CPR; c += 512) {
            const int m  = c / CPR;
            const int kc = (c - m * CPR) * 4; // starting k of this chunk
            const float* src;
            if (SPLIT) {
                src = (kc < H_) ? &A0[(size_t)(row0 + m) * H_ + kc]
                                : &A1[(size_t)(row0 + m) * H_ + (kc - H_)];
            } else {
                src = &A0[(size_t)(row0 + m) * K + kc];
            }
            __builtin_amdgcn_global_load_async_to_lds_b128(
                (GLOBAL_AS v4i*)src, (LDS_AS v4i*)&As[m][kc], 0, 0);
        }
#if __has_builtin(__builtin_amdgcn_s_wait_asynccnt)
        __builtin_amdgcn_s_wait_asynccnt(0);
#else
        asm volatile("s_wait_asynccnt 0x0" ::: "memory");
#endif
    }
#else
    for (int i = tid; i < 16 * K; i += 512) {
        const int m = i / K;
        const int k = i - m * K;
        float v;
        if (SPLIT) {
            v = (k < H_) ? A0[(size_t)(row0 + m) * H_ + k]
                         : A1[(size_t)(row0 + m) * H_ + (k - H_)];
        } else {
            v = A0[(size_t)(row0 + m) * K + k];
        }
        As[m][k] = v;
    }
#endif
    __syncthreads();

    const int wave = tid >> 5;     // 0..15 -> N tile
    const int lane = tid & 31;
    const int half = lane >> 4;    // 0: K pair 0/1, 1: K pair 2/3
    const int l16  = lane & 15;
    const int n0   = wave * 16;

    // 16x16 fp32 accumulator, K stepped by 4 per V_WMMA_F32_16X16X4_F32.
    v8f acc = {};
    for (int k = 0; k < K; k += 4) {
        const int ka = k + 2 * half;
        v2f a, bfrag;
        a.x     = As[l16][ka];
        a.y     = As[l16][ka + 1];
        bfrag.x = W[(size_t)ka * 256 + n0 + l16];
        bfrag.y = W[(size_t)(ka + 1) * 256 + n0 + l16];
        acc = __builtin_amdgcn_wmma_f32_16x16x4_f32(
            /*neg_a=*/false, a, /*neg_b=*/false, bfrag,
            /*c_mod=*/(short)0, acc, /*reuse_a=*/false, /*reuse_b=*/false);
    }

    // Spill accumulator (+bias) to LDS: VGPR r holds M = r + 8*half, N = n0+l16.
#pragma unroll
    for (int r = 0; r < 8; ++r) {
        const int m = r + 8 * half;
        const int n = n0 + l16;
        sm[m][n] = acc[r] + bias[n];
    }
    __syncthreads();

    // GELU (exact, erf) + LayerNorm: wave w owns row w; 8 elements per lane.
    {
        const int m = wave;
        float s = 0.f, ss = 0.f;
        float vals[8];
#pragma unroll
        for (int j = 0; j < 8; ++j) {
            const int n = lane * 8 + j;
            const float x = sm[m][n];
            const float g = 0.5f * x * (1.0f + erff(x * 0.7071067811865475f));
            vals[j] = g;
            s  += g;
            ss += g * g;
        }
#pragma unroll
        for (int o = 16; o > 0; o >>= 1) {
            s  += __shfl_xor(s,  o, 32);
            ss += __shfl_xor(ss, o, 32);
        }
        const float mu   = s * (1.0f / 256.0f);
        const float var  = ss * (1.0f / 256.0f) - mu * mu;
        const float rstd = rsqrtf(var + 1e-5f);
#pragma unroll
        for (int j = 0; j < 8; ++j) {
            const int n = lane * 8 + j;
            out[(size_t)(row0 + m) * 256 + n] = (vals[j] - mu) * rstd * gamma[n] + beta[n];
        }
    }
}

// ---------------------------------------------------------------------------
// Kernel 3: zero scratch (upd accumulator + counts)
// ---------------------------------------------------------------------------
__global__ __launch_bounds__(256)
void zero_kernel(float* __restrict__ p, long long n)
{
    long long i = (long long)blockIdx.x * blockDim.x + threadIdx.x;
    if (i < n) p[i] = 0.f;
}

// ---------------------------------------------------------------------------
// Kernel 4: masked scatter-add of edge features to member nodes + counts.
// ---------------------------------------------------------------------------
__global__ __launch_bounds__(256)
void scatter_kernel(const float* __restrict__ ef,        // [B*E, H]
                    const int* __restrict__ members,     // [B,E,A]
                    const unsigned char* __restrict__ mask,
                    float* __restrict__ upd,             // [B*N, H]
                    float* __restrict__ cnt)             // [B*N]
{
    const int be = blockIdx.x;
    const int b  = be / E_;
    const int h  = threadIdx.x;
    const int* mem = members + (size_t)be * A_;
    const unsigned char* mk = mask + (size_t)be * A_;
    const float v = ef[(size_t)be * H_ + h];

#pragma unroll
    for (int a = 0; a < A_; ++a) {
        if (mk[a]) {
            int idx = mem[a];
            idx = idx < 0 ? 0 : (idx > N_ - 1 ? N_ - 1 : idx);
            atomicAdd(&upd[((size_t)b * N_ + idx) * H_ + h], v);
        }
    }
    if (h < A_) {
        if (mk[h]) {
            int idx = mem[h];
            idx = idx < 0 ? 0 : (idx > N_ - 1 ? N_ - 1 : idx);
            atomicAdd(&cnt[(size_t)b * N_ + idx], 1.0f);
        }
    }
}

// ---------------------------------------------------------------------------
// Kernel 5: normalize scattered sums by counts (clip >= 1).
// ---------------------------------------------------------------------------
__global__ __launch_bounds__(256)
void normalize_kernel(float* __restrict__ upd, const float* __restrict__ cnt)
{
    long long i = (long long)blockIdx.x * blockDim.x + threadIdx.x;
    const long long total = (long long)B_ * N_ * H_;
    if (i < total) {
        const long long row = i / H_;
        upd[i] /= fmaxf(cnt[row], 1.f);
    }
}

// ---------------------------------------------------------------------------
// Host-side orchestration
// ---------------------------------------------------------------------------
extern "C" void kernel_launch(void* const* d_in, const int* in_sizes, int n_in,
                              void* d_out, int out_size, void* d_ws, size_t ws_size,
                              hipStream_t stream)
{
    (void)in_sizes; (void)n_in; (void)out_size; (void)ws_size;

    const float*         nf      = (const float*)d_in[0];          // [B,N,H]
    const int*           members = (const int*)d_in[1];            // [B,E,A]
    const int*           types   = (const int*)d_in[2];            // [B,E]
    const unsigned char* mask    = (const unsigned char*)d_in[3];  // [B,E,A] bool
    const float*         ett     = (const float*)d_in[4];          // [NT,TE]
    const float*         W_enc   = (const float*)d_in[5];          // [288,256]
    const float*         b_enc   = (const float*)d_in[6];
    const float*         g_enc   = (const float*)d_in[7];
    const float*         be_enc  = (const float*)d_in[8];
    const float*         W_upd   = (const float*)d_in[9];          // [512,256]
    const float*         b_upd   = (const float*)d_in[10];
    const float*         g_upd   = (const float*)d_in[11];
    const float*         be_upd  = (const float*)d_in[12];
    float* out = (float*)d_out;

    // Workspace layout (floats); all offsets 16-byte aligned.
    float* ws = (float*)d_ws;
    const size_t off_xin  = 0;                                   // [B*E, 288]
    const size_t off_ef   = off_xin + (size_t)B_ * E_ * KENC;    // [B*E, 256]
    const size_t off_upd  = off_ef  + (size_t)B_ * E_ * H_;      // [B*N, 256]
    const size_t off_cnt  = off_upd + (size_t)B_ * N_ * H_;      // [B*N]
    float* xin = ws + off_xin;
    float* ef  = ws + off_ef;
    float* upd = ws + off_upd;
    float* cnt = ws + off_cnt;

    const int be_blocks  = B_ * E_;                    // 40000
    const int row_blocks = (B_ * E_) / 16;             // 2500 (== B*N/16)

    // 1) pool + embedding concat
    pool_kernel<<<be_blocks, 256, 0, stream>>>(nf, members, types, mask, ett, xin);

    // 2) encoder GEMM + GELU + LN  (WMMA fp32, K=288)
    wmma_gemm_gelu_ln<KENC, false><<<row_blocks, 512, 0, stream>>>(
        xin, nullptr, W_enc, b_enc, g_enc, be_enc, ef);

    // 3) zero scatter accumulators (upd + cnt are contiguous in ws)
    {
        const long long nzero = (long long)B_ * N_ * H_ + (long long)B_ * N_;
        const int blocks = (int)((nzero + 255) / 256);
        zero_kernel<<<blocks, 256, 0, stream>>>(upd, nzero);
    }

    // 4) masked scatter-add
    scatter_kernel<<<be_blocks, 256, 0, stream>>>(ef, members, mask, upd, cnt);

    // 5) normalize by counts
    {
        const long long total = (long long)B_ * N_ * H_;
        const int blocks = (int)((total + 255) / 256);
        normalize_kernel<<<blocks, 256, 0, stream>>>(upd, cnt);
    }

    // 6) update GEMM + GELU + LN (WMMA fp32, K=512, split-A concat on the fly)
    wmma_gemm_gelu_ln<KUPD, true><<<row_blocks, 512, 0, stream>>>(
        nf, upd, W_upd, b_upd, g_upd, be_upd, out);
}